// implicitIterativeShiftModule_47021301957203
// MI455X (gfx1250) — compile-verified
//
#include <hip/hip_runtime.h>

// ---------- types for WMMA ----------
typedef float v2f __attribute__((ext_vector_type(2)));
typedef float v8f __attribute__((ext_vector_type(8)));

#define CG_TOL 1e-5f
#define BT 256   // 8 wave32s per block

// ---------------------------------------------------------------------------
// Exact 32-lane f32 sum using V_WMMA_F32_16X16X4_F32.
// A layout (16x4 f32): VGPR0 = {lanes0-15:K0, lanes16-31:K2}, VGPR1 = {K1,K3}.
// Put s in A-VGPR0, 0 in A-VGPR1  ->  A[m,0]=s_m, A[m,2]=s_{m+16}, rest 0.
// B = all ones -> C[m,n] = s_m + s_{m+16}.  Sum the 8 C VGPRs per lane:
// lanes 0-15 hold rows 0-7, lanes 16-31 rows 8-15; one shfl_xor(16) finishes.
// EXEC must be all ones at the call site (all callers guarantee this).
// ---------------------------------------------------------------------------
__device__ __forceinline__ float wave_reduce32(float s) {
    v2f a; a[0] = s;    a[1] = 0.0f;
    v2f b; b[0] = 1.0f; b[1] = 1.0f;
    v8f c = {};
    c = __builtin_amdgcn_wmma_f32_16x16x4_f32(
        /*neg_a=*/false, a, /*neg_b=*/false, b,
        /*c_mod=*/(short)0, c, /*reuse_a=*/false, /*reuse_b=*/false);
    float t = ((c[0] + c[1]) + (c[2] + c[3])) + ((c[4] + c[5]) + (c[6] + c[7]));
    t += __shfl_xor(t, 16, 32);
    return t;   // all lanes hold the full 32-lane sum
}

// Deterministic block reduction (blockDim.x == BT). Valid on thread 0.
__device__ __forceinline__ float block_reduce(float s, float* lds) {
    float w = wave_reduce32(s);
    int wid  = threadIdx.x >> 5;
    int lane = threadIdx.x & 31;
    if (lane == 0) lds[wid] = w;
    __syncthreads();
    float total = 0.0f;
    if (threadIdx.x == 0) {
        int nw = blockDim.x >> 5;
        for (int k = 0; k < nw; ++k) total += lds[k];   // fixed order
    }
    return total;
}

// ---------------------------------------------------------------------------
// CSR row pointers from sorted i:  rowStart[r] = lower_bound(i, r)
// ---------------------------------------------------------------------------
__global__ void k_rowptr(const int* __restrict__ I, int E,
                         int* __restrict__ rowStart, int n) {
    int r = blockIdx.x * blockDim.x + threadIdx.x;
    if (r > n) return;
    int lo = 0, hi = E;
    while (lo < hi) {
        int mid = (lo + hi) >> 1;
        if (I[mid] < r) lo = mid + 1; else hi = mid;
    }
    rowStart[r] = lo;
}

// ---------------------------------------------------------------------------
// Block-sparse matvec: y[row] = sum_e H_e @ x[j_e], one wave32 per row.
// Lanes read consecutive float4 H blocks (coalesced b128); x gathered as one
// aligned float2 (b64) per edge; gathers hit the 192MB L2 (x is 1.6MB).
// No atomics: deterministic segment sum (fixed butterfly order).
// ---------------------------------------------------------------------------
__global__ void k_matvec(const float4* __restrict__ H, const int* __restrict__ J,
                         const float2* __restrict__ x2, const int* __restrict__ rowStart,
                         float* __restrict__ y, int n) {
    int gtid = blockIdx.x * blockDim.x + threadIdx.x;
    int lane = threadIdx.x & 31;
    int wid  = gtid >> 5;
    int nw   = (gridDim.x * blockDim.x) >> 5;
    for (int row = wid; row < n; row += nw) {
        int s = rowStart[row], e = rowStart[row + 1];
        float ax = 0.0f, ay = 0.0f;
        for (int k = s + lane; k < e; k += 32) {
            __builtin_prefetch(&H[k + 32], 0, 0);   // global_prefetch_b8 (speculative)
            float4 h  = H[k];
            float2 xv = x2[J[k]];
            ax = fmaf(h.x, xv.x, fmaf(h.y, xv.y, ax));
            ay = fmaf(h.z, xv.x, fmaf(h.w, xv.y, ay));
        }
        #pragma unroll
        for (int m = 16; m >= 1; m >>= 1) {
            ax += __shfl_xor(ax, m, 32);
            ay += __shfl_xor(ay, m, 32);
        }
        if (lane == 0) { y[2 * row] = ax; y[2 * row + 1] = ay; }
    }
}

// r = y - B; p = -r; x = x0   (float4 main loop + scalar tail)
__global__ void k_init(const float* __restrict__ y, const float* __restrict__ Bv,
                       const float* __restrict__ x0, float* __restrict__ r,
                       float* __restrict__ p, float* __restrict__ x, int n2) {
    int gid = blockIdx.x * blockDim.x + threadIdx.x;
    int stride = gridDim.x * blockDim.x;
    int n4 = n2 >> 2;
    const float4* y4  = (const float4*)y;
    const float4* B4  = (const float4*)Bv;
    const float4* x04 = (const float4*)x0;
    float4* r4 = (float4*)r; float4* p4 = (float4*)p; float4* x4 = (float4*)x;
    for (int t = gid; t < n4; t += stride) {
        float4 yv = y4[t], bv = B4[t];
        float4 rv = make_float4(yv.x - bv.x, yv.y - bv.y, yv.z - bv.z, yv.w - bv.w);
        r4[t] = rv;
        p4[t] = make_float4(-rv.x, -rv.y, -rv.z, -rv.w);
        x4[t] = x04[t];
    }
    for (int t = (n4 << 2) + gid; t < n2; t += stride) {
        float rv = y[t] - Bv[t];
        r[t] = rv; p[t] = -rv; x[t] = x0[t];
    }
}

// partial dot(a,b) per block -> partials[blockIdx]   (b128 streaming loads)
__global__ void k_dot(const float* __restrict__ a, const float* __restrict__ b,
                      float* __restrict__ partials, int n2) {
    __shared__ float lds[BT / 32];
    int gid = blockIdx.x * blockDim.x + threadIdx.x;
    int stride = gridDim.x * blockDim.x;
    int n4 = n2 >> 2;
    const float4* a4 = (const float4*)a;
    const float4* b4 = (const float4*)b;
    float s = 0.0f;
    for (int t = gid; t < n4; t += stride) {
        float4 av = a4[t], bv = b4[t];
        s += av.x * bv.x + av.y * bv.y + av.z * bv.z + av.w * bv.w;
    }
    for (int t = (n4 << 2) + gid; t < n2; t += stride)
        s += a[t] * b[t];
    float tot = block_reduce(s, lds);   // EXEC all-ones here (reconverged)
    if (threadIdx.x == 0) partials[blockIdx.x] = tot;
}

// sc: [0]=rkrk [1]=alpha [2]=beta [3]=active [4]=active_next [5]=newrr
__global__ void k_reduce(const float* __restrict__ partials, int nb,
                         float* __restrict__ sc, int mode) {
    __shared__ float lds[BT / 32];
    float s = 0.0f;
    for (int t = threadIdx.x; t < nb; t += blockDim.x) s += partials[t];
    float tot = block_reduce(s, lds);
    if (threadIdx.x == 0) {
        if (mode == 0) {                  // init: rkrk, active0
            sc[0] = tot;
            sc[3] = (sqrtf(tot) > CG_TOL) ? 1.0f : 0.0f;
        } else if (mode == 1) {           // alpha = rkrk / (p.Ap)
            sc[1] = sc[0] / tot;
        } else {                          // beta; stage newrr & active_next
            sc[2] = tot / sc[0];
            sc[5] = tot;
            sc[4] = (sc[3] != 0.0f && sqrtf(tot) > CG_TOL) ? 1.0f : 0.0f;
        }
    }
}

// fused: if active { x += alpha*p; r += alpha*Ap; } ; partial dot(r,r)
__global__ void k_update(float* __restrict__ x, float* __restrict__ r,
                         const float* __restrict__ p, const float* __restrict__ Ap,
                         const float* __restrict__ sc, float* __restrict__ partials,
                         int n2) {
    __shared__ float lds[BT / 32];
    float active = sc[3];       // uniform -> no divergence at the WMMA
    float alpha  = sc[1];
    int gid = blockIdx.x * blockDim.x + threadIdx.x;
    int stride = gridDim.x * blockDim.x;
    int n4 = n2 >> 2;
    float4* x4 = (float4*)x; float4* r4 = (float4*)r;
    const float4* p4  = (const float4*)p;
    const float4* Ap4 = (const float4*)Ap;
    float s = 0.0f;
    for (int t = gid; t < n4; t += stride) {
        float4 rv = r4[t];
        if (active != 0.0f) {
            float4 xv = x4[t], pv = p4[t], av = Ap4[t];
            xv.x += alpha * pv.x; xv.y += alpha * pv.y;
            xv.z += alpha * pv.z; xv.w += alpha * pv.w;
            rv.x += alpha * av.x; rv.y += alpha * av.y;
            rv.z += alpha * av.z; rv.w += alpha * av.w;
            x4[t] = xv; r4[t] = rv;
        }
        s += rv.x * rv.x + rv.y * rv.y + rv.z * rv.z + rv.w * rv.w;
    }
    for (int t = (n4 << 2) + gid; t < n2; t += stride) {
        float rv = r[t];
        if (active != 0.0f) {
            float xv = x[t] + alpha * p[t];
            rv = rv + alpha * Ap[t];
            x[t] = xv; r[t] = rv;
        }
        s += rv * rv;
    }
    float tot = block_reduce(s, lds);   // EXEC all-ones here (reconverged)
    if (threadIdx.x == 0) partials[blockIdx.x] = tot;
}

// if active(old): p = -r + beta*p   (uses pre-update active, per reference)
__global__ void k_pupdate(const float* __restrict__ r, float* __restrict__ p,
                          const float* __restrict__ sc, int n2) {
    if (sc[3] == 0.0f) return;  // uniform
    float beta = sc[2];
    int gid = blockIdx.x * blockDim.x + threadIdx.x;
    int stride = gridDim.x * blockDim.x;
    int n4 = n2 >> 2;
    const float4* r4 = (const float4*)r;
    float4* p4 = (float4*)p;
    for (int t = gid; t < n4; t += stride) {
        float4 rv = r4[t], pv = p4[t];
        pv.x = -rv.x + beta * pv.x; pv.y = -rv.y + beta * pv.y;
        pv.z = -rv.z + beta * pv.z; pv.w = -rv.w + beta * pv.w;
        p4[t] = pv;
    }
    for (int t = (n4 << 2) + gid; t < n2; t += stride)
        p[t] = -r[t] + beta * p[t];
}

// commit staged scalars: rkrk <- newrr, active <- active_next
__global__ void k_commit(float* __restrict__ sc) {
    if (threadIdx.x == 0) { sc[0] = sc[5]; sc[3] = sc[4]; }
}

// ---------------------------------------------------------------------------
extern "C" void kernel_launch(void* const* d_in, const int* in_sizes, int n_in,
                              void* d_out, int out_size, void* d_ws, size_t ws_size,
                              hipStream_t stream) {
    const float* H  = (const float*)d_in[0];   // [E,2,2]
    const float* Bv = (const float*)d_in[1];   // [2N]
    const float* x0 = (const float*)d_in[2];   // [2N]
    const int*   I  = (const int*)d_in[3];     // [E], sorted
    const int*   J  = (const int*)d_in[4];     // [E]
    float* x = (float*)d_out;

    const int n2 = in_sizes[1];     // 400,000
    const int n  = n2 / 2;          // 200,000 particles
    const int E  = in_sizes[3];     // 6,400,000 edges

    // carve scratch (256B-aligned regions)
    char* ws = (char*)d_ws;
    size_t off = 0;
    auto carve = [&](size_t bytes) -> void* {
        off = (off + 255) & ~(size_t)255;
        void* pp = ws + off;
        off += bytes;
        return pp;
    };
    int*   rowStart = (int*)  carve((size_t)(n + 1) * sizeof(int));
    float* Ap       = (float*)carve((size_t)n2 * sizeof(float));
    float* r        = (float*)carve((size_t)n2 * sizeof(float));
    float* p        = (float*)carve((size_t)n2 * sizeof(float));
    float* partials = (float*)carve(4096 * sizeof(float));
    float* sc       = (float*)carve(64);

    // vectorized elementwise/dot grid: one thread per float4
    int n4 = (n2 + 3) >> 2;
    int nb = (n4 + BT - 1) / BT;
    if (nb > 1024) nb = 1024;               // bounded partials, grid-stride covers rest
    int mvBlocks = (int)(((long long)n * 32 + BT - 1) / BT);  // one wave32 per row
    int rpBlocks = (n + 1 + BT - 1) / BT;

    // ---- setup: CSR rows, r0 = A x0 - B, p0 = -r0, rkrk, active0 ----
    k_rowptr<<<rpBlocks, BT, 0, stream>>>(I, E, rowStart, n);
    k_matvec<<<mvBlocks, BT, 0, stream>>>((const float4*)H, J, (const float2*)x0,
                                          rowStart, Ap, n);
    k_init  <<<nb, BT, 0, stream>>>(Ap, Bv, x0, r, p, x, n2);
    k_dot   <<<nb, BT, 0, stream>>>(r, r, partials, n2);
    k_reduce<<<1,  BT, 0, stream>>>(partials, nb, sc, 0);

    // ---- 32 masked CG iterations (exact reference where(active,...) semantics) ----
    for (int it = 0; it < 32; ++it) {
        k_matvec <<<mvBlocks, BT, 0, stream>>>((const float4*)H, J, (const float2*)p,
                                               rowStart, Ap, n);
        k_dot    <<<nb, BT, 0, stream>>>(p, Ap, partials, n2);
        k_reduce <<<1,  BT, 0, stream>>>(partials, nb, sc, 1);   // alpha
        k_update <<<nb, BT, 0, stream>>>(x, r, p, Ap, sc, partials, n2);
        k_reduce <<<1,  BT, 0, stream>>>(partials, nb, sc, 2);   // beta, staged active
        k_pupdate<<<nb, BT, 0, stream>>>(r, p, sc, n2);
        k_commit <<<1,  1,  0, stream>>>(sc);
    }
}